// HybridAttention_65481071407886
// MI455X (gfx1250) — compile-verified
//
#include <hip/hip_runtime.h>

typedef float v2f __attribute__((ext_vector_type(2)));
typedef float v8f __attribute__((ext_vector_type(8)));

#define B_  4
#define S_  512
#define E_  128
#define H_  16
#define DK_ 8
#define M_TOT (B_ * S_)          // 2048 rows
#define KC  32                   // flash-attention key chunk (LDS budget)

// D = A(16x4 f32) * B(4x16 f32) + C(16x16 f32), full fp32 path on gfx1250.
static __device__ __forceinline__ v8f wmma_f32(v2f a, v2f b, v8f c) {
  return __builtin_amdgcn_wmma_f32_16x16x4_f32(
      /*neg_a=*/false, a, /*neg_b=*/false, b,
      /*c_mod=*/(short)0, c, /*reuse_a=*/false, /*reuse_b=*/false);
}

// ---------------------------------------------------------------------------
// Kernel 1: qkv = x @ Win^T + bias, scattered into Q/K/V as [B,H,S,DK].
// Grid: 128 M-tiles x 24 N-tiles = 3072 waves; 8 waves/block -> 384 blocks.
// ---------------------------------------------------------------------------
__global__ __launch_bounds__(256) void qkv_gemm(
    const float* __restrict__ x,     // [2048,128]
    const float* __restrict__ w,     // [384,128]  (out = x @ w^T)
    const float* __restrict__ bias,  // [384]
    float* __restrict__ Q, float* __restrict__ K, float* __restrict__ V) {
  const int gw   = (blockIdx.x * 256 + threadIdx.x) >> 5;
  const int lane = threadIdx.x & 31;
  const int half = lane >> 4;        // 0: K rows 0/1, 1: K rows 2/3
  const int l    = lane & 15;
  const int mt = gw / 24, nt = gw - mt * 24;
  const int m0 = mt << 4, n0 = nt << 4;

  const float* xr = x + (size_t)(m0 + l) * E_ + half * 2;
  const float* wr = w + (size_t)(n0 + l) * E_ + half * 2;

  v8f acc = {0.f, 0.f, 0.f, 0.f, 0.f, 0.f, 0.f, 0.f};
#pragma unroll 8
  for (int k = 0; k < E_; k += 4) {
    __builtin_prefetch(wr + k + 32, 0, 1);   // global_prefetch_b8 of weight stream
    v2f a, b;
    a.x = xr[k]; a.y = xr[k + 1];
    b.x = wr[k]; b.y = wr[k + 1];
    acc = wmma_f32(a, b, acc);
  }

  const int n = n0 + l;
  const int which = n >> 7;          // 0:q 1:k 2:v (n0 multiple of 16, so uniform)
  const int e = n & 127;
  const int h = e >> 3, dk = e & 7;
  float* dst = (which == 0) ? Q : (which == 1 ? K : V);
  const float bn = bias[n];
#pragma unroll
  for (int r = 0; r < 8; ++r) {
    const int m  = m0 + r + half * 8;
    const int bb = m >> 9, s = m & 511;
    dst[(((size_t)(bb * H_ + h) * S_) + s) * DK_ + dk] = acc[r] + bn;
  }
}

// ---------------------------------------------------------------------------
// Kernel 2: flash attention per (b,h). 64 blocks, 8 waves each.
// K/V for the whole (b,h) are staged into LDS once per block via the CDNA5
// async-to-LDS path (global_load_async_to_lds_b128, ASYNCcnt), then K, V and P WMMA
// fragments are built from ds_load. Each wave owns q-tiles qt = wv, wv+8, ...
// and a private 16xKC probability slab in LDS.
// LDS: K 16K + V 16K + slabs 16K + stats 1.5K = 49.5 KB.
// ---------------------------------------------------------------------------
__global__ __launch_bounds__(256) void attn_fa(
    const float* __restrict__ Q, const float* __restrict__ K,
    const float* __restrict__ V, float* __restrict__ attn /* [2048,128] */) {
  __shared__ float Ks[S_ * DK_];                   // 16 KB
  __shared__ float Vs[S_ * DK_];                   // 16 KB
  __shared__ float slab[8][16 * KC];               // 16 KB: per-wave P tile
  __shared__ float stat_m[8][16], stat_l[8][16], stat_a[8][16];

  const int bh   = blockIdx.x;                     // 0..63
  const int b    = bh >> 4, h = bh & 15;
  const int wv   = threadIdx.x >> 5;
  const int lane = threadIdx.x & 31;
  const int half = lane >> 4, l = lane & 15;
  const float scale = 0.35355339059327373f;        // 1/sqrt(DK)

  const float* Qb = Q + (size_t)bh * S_ * DK_;
  const float* Kb = K + (size_t)bh * S_ * DK_;
  const float* Vb = V + (size_t)bh * S_ * DK_;
  float* sl = slab[wv];

  // ---- async stage K and V into LDS (b128 per lane, 4 iters per thread) ----
  {
    const unsigned ldsK = (unsigned)(size_t)(&Ks[0]);
    const unsigned ldsV = (unsigned)(size_t)(&Vs[0]);
#pragma unroll
    for (int i = 0; i < (S_ * DK_) / (256 * 4); ++i) {       // 4 iterations
      const int f0 = (i * 256 + (int)threadIdx.x) * 4;       // float index
      const unsigned lk = ldsK + (unsigned)f0 * 4u;
      const unsigned lv = ldsV + (unsigned)f0 * 4u;
      const float* gk = Kb + f0;
      const float* gv = Vb + f0;
      asm volatile("global_load_async_to_lds_b128 %0, %1, off"
                   :: "v"(lk), "v"(gk) : "memory");
      asm volatile("global_load_async_to_lds_b128 %0, %1, off"
                   :: "v"(lv), "v"(gv) : "memory");
    }
    asm volatile("s_wait_asynccnt 0x0" ::: "memory");
    __syncthreads();
  }

  for (int qt = wv; qt < S_ / 16; qt += 8) {
    const int m0 = qt << 4;
    // Q fragments for the two k-steps (dk 0..3 and 4..7)
    const float* qrow = Qb + (size_t)(m0 + l) * DK_ + half * 2;
    v2f qa0, qa1;
    qa0.x = qrow[0]; qa0.y = qrow[1];
    qa1.x = qrow[4]; qa1.y = qrow[5];

    v8f o = {0.f, 0.f, 0.f, 0.f, 0.f, 0.f, 0.f, 0.f};
    if (lane < 16) { stat_m[wv][l] = -1e30f; stat_l[wv][l] = 0.f; }

    for (int c0 = 0; c0 < S_; c0 += KC) {
      // ---- scores for KC keys: (KC/16) subtiles x (2 WMMA k-steps) ----
#pragma unroll
      for (int st = 0; st < KC / 16; ++st) {
        const int kk = c0 + st * 16;
        const float* krow = &Ks[(kk + l) * DK_ + half * 2];
        v2f kb0, kb1;
        kb0.x = krow[0]; kb0.y = krow[1];
        kb1.x = krow[4]; kb1.y = krow[5];
        v8f s = {0.f, 0.f, 0.f, 0.f, 0.f, 0.f, 0.f, 0.f};
        s = wmma_f32(qa0, kb0, s);
        s = wmma_f32(qa1, kb1, s);
#pragma unroll
        for (int r = 0; r < 8; ++r)
          sl[(r + half * 8) * KC + st * 16 + l] = s[r] * scale;
      }

      // ---- online softmax: two lanes (half=0/1) cooperate per row l ----
      const int j0 = half * (KC / 2);
      float mx = -1e30f;
#pragma unroll
      for (int j = 0; j < KC / 2; ++j) mx = fmaxf(mx, sl[l * KC + j0 + j]);
      mx = fmaxf(mx, __shfl_xor(mx, 16, 32));
      const float mold  = stat_m[wv][l];
      const float mnew  = fmaxf(mold, mx);
      const float alpha = __expf(mold - mnew);     // 0 on first chunk
      float psum = 0.f;
#pragma unroll
      for (int j = 0; j < KC / 2; ++j) {
        const float p = __expf(sl[l * KC + j0 + j] - mnew);
        sl[l * KC + j0 + j] = p;
        psum += p;
      }
      psum += __shfl_xor(psum, 16, 32);
      if (lane < 16) {
        stat_m[wv][l] = mnew;
        stat_l[wv][l] = stat_l[wv][l] * alpha + psum;
        stat_a[wv][l] = alpha;
      }

      // ---- rescale running output by per-row alpha ----
#pragma unroll
      for (int r = 0; r < 8; ++r) o[r] *= stat_a[wv][r + half * 8];

      // ---- O += P(16xKC) @ V(KCx8), V padded to N=16, all from LDS ----
#pragma unroll
      for (int st = 0; st < KC / 4; ++st) {
        const int k4 = st * 4;
        v2f pa, vb;
        pa.x = sl[l * KC + k4 + half * 2 + 0];
        pa.y = sl[l * KC + k4 + half * 2 + 1];
        const int key = c0 + k4 + half * 2;
        const bool ok = (l < DK_);
        vb.x = ok ? Vs[(key + 0) * DK_ + l] : 0.f;
        vb.y = ok ? Vs[(key + 1) * DK_ + l] : 0.f;
        o = wmma_f32(pa, vb, o);
      }
    }

    // ---- normalize and store into [B,S,E] attention buffer ----
#pragma unroll
    for (int r = 0; r < 8; ++r) {
      const int row = r + half * 8;
      const float inv = 1.0f / stat_l[wv][row];
      if (l < DK_)
        attn[(((size_t)b * S_) + m0 + row) * E_ + h * DK_ + l] = o[r] * inv;
    }
  }
}

// ---------------------------------------------------------------------------
// Kernel 3: out = (1-g) * (attn @ Wo^T + bo) + g * qh(x, theta)
// qh from the analytic collapse of the RX/CNOT-ring circuit:
//   phi_j = x_j + theta_j ; w==0: prod_{j=1..7} cos(phi_j) ; w>=1: prod_{j=0..w}
// Grid: 128 M-tiles x 8 N-tiles = 1024 waves -> 128 blocks.
// ---------------------------------------------------------------------------
__global__ __launch_bounds__(256) void out_proj_quantum(
    const float* __restrict__ x,     // [2048,128] raw input (for angles)
    const float* __restrict__ attn,  // [2048,128]
    const float* __restrict__ wo,    // [128,128]
    const float* __restrict__ bo,    // [128]
    const float* __restrict__ theta, // [16,8]
    const float* __restrict__ gate,  // [1]
    float* __restrict__ out) {
  const int gw   = (blockIdx.x * 256 + threadIdx.x) >> 5;
  const int lane = threadIdx.x & 31;
  const int half = lane >> 4, l = lane & 15;
  const int mt = gw >> 3, nt = gw & 7;
  const int m0 = mt << 4, n0 = nt << 4;

  const float* ar = attn + (size_t)(m0 + l) * E_ + half * 2;
  const float* wr = wo   + (size_t)(n0 + l) * E_ + half * 2;
  v8f acc = {0.f, 0.f, 0.f, 0.f, 0.f, 0.f, 0.f, 0.f};
#pragma unroll 8
  for (int k = 0; k < E_; k += 4) {
    v2f a, b;
    a.x = ar[k]; a.y = ar[k + 1];
    b.x = wr[k]; b.y = wr[k + 1];
    acc = wmma_f32(a, b, acc);
  }

  const float g = 1.f / (1.f + __expf(-gate[0]));
  const int n = n0 + l;
  const int h = n >> 3, w = n & 7;
  float th[8];
#pragma unroll
  for (int j = 0; j < 8; ++j) th[j] = theta[h * 8 + j];
  const float bn  = bo[n];
  const int  jlo = (w == 0) ? 1 : 0;
  const int  jhi = (w == 0) ? 7 : w;

#pragma unroll
  for (int r = 0; r < 8; ++r) {
    const int m = m0 + r + half * 8;
    const float* xr = x + (size_t)m * E_ + h * DK_;
    float prod = 1.f;
    for (int j = jlo; j <= jhi; ++j)
      prod *= __cosf(xr[j] + th[j]);
    out[(size_t)m * E_ + n] = (1.f - g) * (acc[r] + bn) + g * prod;
  }
}

// ---------------------------------------------------------------------------
extern "C" void kernel_launch(void* const* d_in, const int* in_sizes, int n_in,
                              void* d_out, int out_size, void* d_ws, size_t ws_size,
                              hipStream_t stream) {
  const float* x  = (const float*)d_in[0];
  const float* wi = (const float*)d_in[1];
  const float* bi = (const float*)d_in[2];
  const float* wo = (const float*)d_in[3];
  const float* bo = (const float*)d_in[4];
  const float* th = (const float*)d_in[5];
  const float* gt = (const float*)d_in[6];
  float* out = (float*)d_out;

  // workspace layout (floats): Q | K | V  each [B,H,S,DK] = 262144, attn [2048,128]
  float* ws = (float*)d_ws;
  float* Q = ws;
  float* K = ws + 262144;
  float* V = ws + 524288;
  float* A = ws + 786432;                  // total 4 MB of scratch

  qkv_gemm<<<384, 256, 0, stream>>>(x, wi, bi, Q, K, V);
  attn_fa<<<64, 256, 0, stream>>>(Q, K, V, A);
  out_proj_quantum<<<128, 256, 0, stream>>>(x, A, wo, bo, th, gt, out);
}